// SAGE_36661840838929
// MI455X (gfx1250) — compile-verified
//
#include <hip/hip_runtime.h>

// ---------------------------------------------------------------------------
// GraphSAGE 2-layer forward on MI455X (gfx1250), wave32 + WMMA f32 16x16x4.
//   L1: mean-aggr(x) @ W1l.T + b1l + x @ W1r.T -> L2norm -> ReLU -> h [N,32]
//   L2: mean-aggr(h) @ W2l.T + b2l + h @ W2r.T -> L2norm -> log_softmax [N,16]
// ---------------------------------------------------------------------------

#define N_NODES 100000
#define N_EDGES 1600000
#define IN_CH   50
#define HID_CH  32
#define OUT_CH  16
#define NTILES  (N_NODES / 16)   // 6250, exact

typedef __attribute__((ext_vector_type(2))) float v2f;
typedef __attribute__((ext_vector_type(8))) float v8f;

__device__ __forceinline__ v8f wmma_f32x4(v2f a, v2f b, v8f c) {
    // V_WMMA_F32_16X16X4_F32 : D = A(16x4) * B(4x16) + C(16x16), all f32.
    // 8-arg form: (neg_a, A, neg_b, B, c_mod, C, reuse_a, reuse_b)
    return __builtin_amdgcn_wmma_f32_16x16x4_f32(
        false, a, false, b, (short)0, c, false, false);
}

// ---------------------------------------------------------------------------
// Zero a float region (grid-stride).
// ---------------------------------------------------------------------------
__global__ void k_zero(float* __restrict__ p, int n) {
    int i = blockIdx.x * blockDim.x + threadIdx.x;
    int stride = gridDim.x * blockDim.x;
    for (; i < n; i += stride) p[i] = 0.0f;
}

// ---------------------------------------------------------------------------
// Layer-1 scatter: 64 threads per edge (50 active channels).
// aggr[dst, c] += x[src, c];  deg[dst] += 1 (channel 0 only).
// ---------------------------------------------------------------------------
__global__ void k_scatter1(const float* __restrict__ x,
                           const int* __restrict__ ei,
                           float* __restrict__ aggr,
                           float* __restrict__ deg) {
    unsigned id = blockIdx.x * blockDim.x + threadIdx.x;
    unsigned e = id >> 6;
    unsigned c = id & 63u;
    if (e >= N_EDGES) return;
    int s = ei[e];
    int d = ei[N_EDGES + e];
    if (c < IN_CH) atomicAdd(&aggr[d * IN_CH + (int)c], x[s * IN_CH + (int)c]);
    if (c == 0)    atomicAdd(&deg[d], 1.0f);
}

// ---------------------------------------------------------------------------
// Layer-2 scatter: one wave per edge, 32 channels, fully coalesced.
// aggr2[dst, c] += h[src, c]
// ---------------------------------------------------------------------------
__global__ void k_scatter2(const float* __restrict__ h,
                           const int* __restrict__ ei,
                           float* __restrict__ aggr2) {
    unsigned id = blockIdx.x * blockDim.x + threadIdx.x;
    unsigned e = id >> 5;
    unsigned c = id & 31u;
    if (e >= N_EDGES) return;
    int s = ei[e];
    int d = ei[N_EDGES + e];
    atomicAdd(&aggr2[d * HID_CH + (int)c], h[s * HID_CH + (int)c]);
}

// ---------------------------------------------------------------------------
// Layer-1 GEMM: one wave32 per 16-node tile.
//   out[16,32] = mean[16,50] @ W1l.T + b1l + x[16,50] @ W1r.T
//   then row L2-normalize + ReLU -> h.
// Exact launch geometry (3125 blocks x 64 threads = 6250 waves = NTILES),
// so EXEC is all-1s at every WMMA.
// A-frag (16x4 f32): lane<16 -> K=k0,k0+1 ; lane>=16 -> K=k0+2,k0+3; row = lane&15.
// B-frag (4x16 f32): same K split; col = lane&15.
// C/D (16x16 f32):   vgpr v holds row (v + 8*(lane>=16)), col = lane&15.
// ---------------------------------------------------------------------------
__global__ void k_gemm1(const float* __restrict__ x,
                        const float* __restrict__ aggr,
                        const float* __restrict__ deg,
                        const float* __restrict__ W1l,
                        const float* __restrict__ b1l,
                        const float* __restrict__ W1r,
                        float* __restrict__ h) {
    const int lane = threadIdx.x & 31;
    const int wave = (blockIdx.x * blockDim.x + threadIdx.x) >> 5;  // < NTILES by construction
    const int l16  = lane & 15;
    const int half = lane >> 4;
    const int row0 = wave * 16;
    const int mrow = row0 + l16;

    const float invd = 1.0f / fmaxf(deg[mrow], 1.0f);

    const float* __restrict__ xr  = x    + mrow * IN_CH;
    const float* __restrict__ ar  = aggr + mrow * IN_CH;
    const float* __restrict__ wl0 = W1l + l16 * IN_CH;
    const float* __restrict__ wl1 = W1l + (l16 + 16) * IN_CH;
    const float* __restrict__ wr0 = W1r + l16 * IN_CH;
    const float* __restrict__ wr1 = W1r + (l16 + 16) * IN_CH;

    v8f acc0 = {};  // output cols 0..15
    v8f acc1 = {};  // output cols 16..31

    #pragma unroll
    for (int k0 = 0; k0 < 52; k0 += 4) {          // K padded 50 -> 52
        const int  kb = k0 + 2 * half;
        const bool g0 = kb < IN_CH;
        const bool g1 = (kb + 1) < IN_CH;
        v2f am, ax, bl0, bl1, br0, br1;
        am.x  = g0 ? ar[kb] * invd : 0.0f;  am.y  = g1 ? ar[kb + 1] * invd : 0.0f;
        ax.x  = g0 ? xr[kb]        : 0.0f;  ax.y  = g1 ? xr[kb + 1]        : 0.0f;
        bl0.x = g0 ? wl0[kb]       : 0.0f;  bl0.y = g1 ? wl0[kb + 1]       : 0.0f;
        bl1.x = g0 ? wl1[kb]       : 0.0f;  bl1.y = g1 ? wl1[kb + 1]       : 0.0f;
        br0.x = g0 ? wr0[kb]       : 0.0f;  br0.y = g1 ? wr0[kb + 1]       : 0.0f;
        br1.x = g0 ? wr1[kb]       : 0.0f;  br1.y = g1 ? wr1[kb + 1]       : 0.0f;
        acc0 = wmma_f32x4(am, bl0, acc0);
        acc0 = wmma_f32x4(ax, br0, acc0);
        acc1 = wmma_f32x4(am, bl1, acc1);
        acc1 = wmma_f32x4(ax, br1, acc1);
    }

    const float bias0 = b1l[l16];
    const float bias1 = b1l[l16 + 16];

    #pragma unroll
    for (int v = 0; v < 8; ++v) {
        float o0 = acc0[v] + bias0;
        float o1 = acc1[v] + bias1;
        // Row L2 norm: this row lives across the 16 lanes of this half-wave.
        float s = o0 * o0 + o1 * o1;
        s += __shfl_xor(s, 1);
        s += __shfl_xor(s, 2);
        s += __shfl_xor(s, 4);
        s += __shfl_xor(s, 8);
        const float inv = 1.0f / fmaxf(sqrtf(s), 1e-12f);
        const int m = row0 + v + 8 * half;
        h[m * HID_CH + l16]      = fmaxf(o0 * inv, 0.0f);  // ReLU
        h[m * HID_CH + 16 + l16] = fmaxf(o1 * inv, 0.0f);
    }
}

// ---------------------------------------------------------------------------
// Layer-2 GEMM: one wave32 per 16-node tile, K=32, single 16-wide N tile.
//   out = mean2 @ W2l.T + b2l + h @ W2r.T -> L2 normalize -> log_softmax.
// ---------------------------------------------------------------------------
__global__ void k_gemm2(const float* __restrict__ h,
                        const float* __restrict__ aggr2,
                        const float* __restrict__ deg,
                        const float* __restrict__ W2l,
                        const float* __restrict__ b2l,
                        const float* __restrict__ W2r,
                        float* __restrict__ out) {
    const int lane = threadIdx.x & 31;
    const int wave = (blockIdx.x * blockDim.x + threadIdx.x) >> 5;
    const int l16  = lane & 15;
    const int half = lane >> 4;
    const int row0 = wave * 16;
    const int mrow = row0 + l16;

    const float invd = 1.0f / fmaxf(deg[mrow], 1.0f);

    const float* __restrict__ hr = h     + mrow * HID_CH;
    const float* __restrict__ ar = aggr2 + mrow * HID_CH;
    const float* __restrict__ wl = W2l + l16 * HID_CH;
    const float* __restrict__ wr = W2r + l16 * HID_CH;

    v8f acc = {};
    #pragma unroll
    for (int k0 = 0; k0 < HID_CH; k0 += 4) {
        const int kb = k0 + 2 * half;
        v2f am = { ar[kb] * invd, ar[kb + 1] * invd };
        v2f ah = { hr[kb],        hr[kb + 1]        };
        v2f bl = { wl[kb],        wl[kb + 1]        };
        v2f br = { wr[kb],        wr[kb + 1]        };
        acc = wmma_f32x4(am, bl, acc);
        acc = wmma_f32x4(ah, br, acc);
    }

    const float bias = b2l[l16];

    #pragma unroll
    for (int v = 0; v < 8; ++v) {
        float o = acc[v] + bias;
        // L2 normalize this 16-wide row (across 16 lanes of this half-wave).
        float s = o * o;
        s += __shfl_xor(s, 1);
        s += __shfl_xor(s, 2);
        s += __shfl_xor(s, 4);
        s += __shfl_xor(s, 8);
        float val = o * (1.0f / fmaxf(sqrtf(s), 1e-12f));
        // log_softmax over the 16 channels.
        float mx = val;
        mx = fmaxf(mx, __shfl_xor(mx, 1));
        mx = fmaxf(mx, __shfl_xor(mx, 2));
        mx = fmaxf(mx, __shfl_xor(mx, 4));
        mx = fmaxf(mx, __shfl_xor(mx, 8));
        float ex = expf(val - mx);
        float se = ex;
        se += __shfl_xor(se, 1);
        se += __shfl_xor(se, 2);
        se += __shfl_xor(se, 4);
        se += __shfl_xor(se, 8);
        const int m = row0 + v + 8 * half;
        out[m * OUT_CH + l16] = val - mx - logf(se);
    }
}

// ---------------------------------------------------------------------------
// Host-side launch. Workspace layout (floats):
//   [0, N)                : deg
//   [N, N + N*50)         : aggr1  (layer-2 reuses first N*32 as aggr2)
//   [N + N*50, +N*32)     : h
// Total = N*(1+50+32)*4 bytes ~= 33.2 MB.
// ---------------------------------------------------------------------------
extern "C" void kernel_launch(void* const* d_in, const int* in_sizes, int n_in,
                              void* d_out, int out_size, void* d_ws, size_t ws_size,
                              hipStream_t stream) {
    const float* x   = (const float*)d_in[0];
    const int*   ei  = (const int*)d_in[1];   // [2, E]: src rows then dst rows
    const float* W1l = (const float*)d_in[2];
    const float* b1l = (const float*)d_in[3];
    const float* W1r = (const float*)d_in[4];
    const float* W2l = (const float*)d_in[5];
    const float* b2l = (const float*)d_in[6];
    const float* W2r = (const float*)d_in[7];
    float* out = (float*)d_out;

    float* ws    = (float*)d_ws;
    float* deg   = ws;                                    // N
    float* aggr1 = ws + N_NODES;                          // N*50 (reused as aggr2)
    float* h     = ws + N_NODES + N_NODES * IN_CH;        // N*32

    // Phase 0: zero deg + aggr1 (contiguous).
    {
        int n = N_NODES * (1 + IN_CH);
        k_zero<<<(n + 255) / 256, 256, 0, stream>>>(ws, n);
    }
    // Phase 1: edge scatter for layer 1 (x -> aggr1, deg).
    {
        unsigned total = (unsigned)N_EDGES * 64u;
        k_scatter1<<<(total + 255u) / 256u, 256, 0, stream>>>(x, ei, aggr1, deg);
    }
    // Phase 2: layer-1 WMMA GEMM + L2norm + ReLU -> h. Exact geometry, no guards.
    k_gemm1<<<NTILES / 2, 64, 0, stream>>>(x, aggr1, deg, W1l, b1l, W1r, h);

    // Phase 3: zero aggr2 (reuse of aggr1's first N*32 floats).
    {
        int n = N_NODES * HID_CH;
        k_zero<<<(n + 255) / 256, 256, 0, stream>>>(aggr1, n);
    }
    // Phase 4: edge scatter for layer 2 (h -> aggr2).
    {
        unsigned total = (unsigned)N_EDGES * 32u;
        k_scatter2<<<(total + 255u) / 256u, 256, 0, stream>>>(h, ei, aggr1);
    }
    // Phase 5: layer-2 WMMA GEMM + L2norm + log_softmax -> out.
    k_gemm2<<<NTILES / 2, 64, 0, stream>>>(h, aggr1, deg, W2l, b2l, W2r, out);
}